// rGIN_27882927686091
// MI455X (gfx1250) — compile-verified
//
#include <hip/hip_runtime.h>

// rGIN aggregation: out = h + segment_sum(h[src], dst), h = [nodes | random]
// N=100000, F=128, M=1600000.  Memory/atomic bound; L2-resident gathers+atomics.
// CDNA5 path: TDM tensor_load_to_lds stages the 2-row edge-index tile into LDS.

typedef __attribute__((ext_vector_type(4))) unsigned int v4u;
typedef __attribute__((ext_vector_type(8))) int          v8i;
typedef __attribute__((ext_vector_type(4))) int          v4i;

#define TILE 1024   // edges per workgroup tile (fits 16-bit tile_dim0)
#define FEAT 128

// ---------------- Kernel: scatter-add h[src] into out[dst] ------------------
__global__ __launch_bounds__(256) void rgin_scatter(const float* __restrict__ nodes,
                                                    const int*   __restrict__ eidx,
                                                    const float* __restrict__ rnd,
                                                    float* __restrict__ out,
                                                    int M) {
    __shared__ int s_idx[2 * TILE];   // [0:TILE) = dst row, [TILE:2*TILE) = src row

    const int e0 = blockIdx.x * TILE;

    // Wave 0 issues one TDM load: 2D tile (TILE x 2) of int32, row stride = M
    // elements, i.e. row0 = edge_index[0] (dst), row1 = edge_index[1] (src).
    if ((threadIdx.x >> 5) == 0) {
        unsigned long long ga = (unsigned long long)(const void*)(eidx + e0);
        unsigned int lds_addr = (unsigned int)(unsigned long long)(void*)s_idx;
        unsigned int td0 = (unsigned int)(M - e0);   // remaining dim0 for OOB

        // D# group 0: count=1 | lds_addr | global_addr[56:0] | type=2
        v4u g0;
        g0.x = 1u;                                     // count=1, user mode
        g0.y = lds_addr;                               // LDS byte address
        g0.z = (unsigned int)ga;                       // global_addr[31:0]
        g0.w = (unsigned int)((ga >> 32) & 0x1FFFFFFu) // global_addr[56:32]
               | (2u << 30);                           // type = 2 (image)

        // D# group 1: data_size=4B, tensor_dim0=td0, tensor_dim1=2,
        // tile_dim0=TILE, tile_dim1=2, tensor_dim0_stride=M
        v8i g1;
        g1[0] = (int)0x00020000u;                            // data_size=2 (4B)
        g1[1] = (int)((td0 & 0xFFFFu) << 16);                // tensor_dim0 lo16
        g1[2] = (int)(((td0 >> 16) & 0xFFFFu) | (2u << 16)); // td0 hi | tensor_dim1=2
        g1[3] = (int)((unsigned int)TILE << 16);             // td1 hi=0 | tile_dim0
        g1[4] = 2;                                           // tile_dim1=2, tile_dim2=0
        g1[5] = M;                                           // dim0_stride[31:0]
        g1[6] = 0;                                           // dim0_stride[47:32], dim1_stride lo
        g1[7] = 0;                                           // dim1_stride hi

        v4i gz4 = {0, 0, 0, 0};                              // groups 2/3: 2D tile only
        v8i gz8 = {0, 0, 0, 0, 0, 0, 0, 0};                  // extra group (clang-23 form)

        __builtin_amdgcn_tensor_load_to_lds(g0, g1, gz4, gz4, gz8, 0);
        __builtin_amdgcn_s_wait_tensorcnt(0);
    }
    __syncthreads();

    const int lane = threadIdx.x & 31;
    const int wave = threadIdx.x >> 5;
    const int rem  = M - e0;
    const int lim  = rem < TILE ? rem : TILE;   // countable loop -> pipelining

    // Each wave handles one edge at a time; 32 lanes x float4 cover 128 feats.
    for (int i = wave; i < lim; i += 8) {
        int dst = s_idx[i];
        int src = s_idx[TILE + i];

        const float4 v = *(const float4*)(nodes + (size_t)src * FEAT + (lane << 2));
        float* o = out + (size_t)dst * (FEAT + 1) + (lane << 2);
        atomicAdd(o + 0, v.x);
        atomicAdd(o + 1, v.y);
        atomicAdd(o + 2, v.z);
        atomicAdd(o + 3, v.w);
        if (lane == 0)
            atomicAdd(out + (size_t)dst * (FEAT + 1) + FEAT, rnd[src]);
    }
}

// ---------------- Kernel: out[n][0:128] = nodes[n], out[n][128] = rnd[n] ----
__global__ __launch_bounds__(256) void rgin_init_h(const float* __restrict__ nodes,
                                                   const float* __restrict__ rnd,
                                                   float* __restrict__ out,
                                                   int N) {
    int i = blockIdx.x * blockDim.x + threadIdx.x;
    int total = N * (FEAT + 1);
    if (i >= total) return;
    int n = i / (FEAT + 1);
    int f = i - n * (FEAT + 1);
    out[i] = (f < FEAT) ? nodes[n * FEAT + f] : rnd[n];
}

extern "C" void kernel_launch(void* const* d_in, const int* in_sizes, int n_in,
                              void* d_out, int out_size, void* d_ws, size_t ws_size,
                              hipStream_t stream) {
    const float* nodes = (const float*)d_in[0];   // [N, 128] f32
    const int*   eidx  = (const int*)d_in[1];     // [2, M] int32 (JAX x64 off)
    const float* rnd   = (const float*)d_in[2];   // [N, 1] f32
    float* out = (float*)d_out;                   // [N, 129] f32

    const int N = in_sizes[2];
    const int M = in_sizes[1] / 2;

    const int total = N * (FEAT + 1);
    rgin_init_h<<<(total + 255) / 256, 256, 0, stream>>>(nodes, rnd, out, N);

    const int nblocks = (M + TILE - 1) / TILE;
    rgin_scatter<<<nblocks, 256, 0, stream>>>(nodes, eidx, rnd, out, M);
}